// SDGraphEncoder_79310866088305
// MI455X (gfx1250) — compile-verified
//
#include <hip/hip_runtime.h>
#include <hip/hip_bf16.h>

typedef __attribute__((ext_vector_type(2))) float v2f;
typedef __attribute__((ext_vector_type(8))) float v8f;

// ---------------------------------------------------------------------------
// Fused GEMM: Y[b] = W (O x K) @ X[b] (K x ncols), fp32 WMMA 16x16x4.
// B-operand MODEs (fused data staging, avoids materializing big tensors):
//   0 = plain            X = Xa[k, col]
//   1 = concat           k < Ca -> Xa, else Xb               (channel concat)
//   2 = concat+broadcast k < Ca -> Xa, else Xb[., col/inner] (sparse->dense)
//   3 = graph feature    k < Ca -> Xa[k,nbr]-Xa[k,cen], else Xa[k-Ca,cen]
//   4 = temporal im2col  j=(c,tau): Xa[c, clamp(col+tau-2)]  (edge-padded k=5)
// Bias is omitted everywhere: every conv feeds training-mode BN, which
// subtracts the per-channel mean, so a constant bias cancels exactly.
//
// Guard-free hot loops: WMMA column n depends only on B column n and row m
// only on A row m, so OOB rows/cols are merely address-clamped (never stored).
// Only the K tail needs zero-fill; it runs as one branchless select iteration.
// ---------------------------------------------------------------------------
template<int MODE>
__global__ __launch_bounds__(128)
void gemm_wmma_k(const float* __restrict__ W,
                 const float* __restrict__ Xa, long sXa, int lda,
                 const float* __restrict__ Xb, long sXb, int ldb,
                 const int* __restrict__ nbr, long sNbr,
                 float* __restrict__ Y, long sY,
                 int O, int K, int Ca, int ncols, int knn, int inner)
{
  const int lane = threadIdx.x;          // 0..31 (wave32)
  const int lm   = lane & 15;
  const int kh   = (lane >> 4) << 1;     // lanes 16-31 hold K+2,K+3 halves
  const int m0   = blockIdx.y * 16;
  const int n0   = (blockIdx.x * 4 + threadIdx.y) * 32;  // 4 waves/block
  if (n0 >= ncols) return;               // uniform per wave
  const int b = blockIdx.z;
  Xa += (long)b * sXa;
  if (MODE == 1 || MODE == 2) Xb += (long)b * sXb;
  Y  += (long)b * sY;

  const int row  = m0 + lm;
  const int rowc = row < O ? row : O - 1;          // clamp, store is guarded
  const int col0 = n0 + lm;                         // ncols is a multiple of 32
  const int col1 = n0 + 16 + lm;
  const int cb0  = (MODE == 2) ? col0 / inner : 0;  // broadcast column
  const int cb1  = (MODE == 2) ? col1 / inner : 0;
  int cen0 = 0, cen1 = 0, nb0 = 0, nb1 = 0;
  if (MODE == 3) {
    const int* ib = nbr + (long)b * sNbr;
    cen0 = col0 / knn;  cen1 = col1 / knn;
    nb0  = ib[col0];    nb1  = ib[col1];
  }
  const int Cb = (MODE == 0 || MODE == 4) ? K : Ca;  // first-region length

  // ---- operand loaders -----------------------------------------------------
  auto ldwf = [&](int k) -> float {                  // A, k known < K
    if (MODE == 4) { int c = k & 127, tau = k >> 7;
                     return W[(long)rowc * K + c * 5 + tau]; }
    return W[(long)rowc * K + k];
  };
  auto ldw = [&](int k) -> float {                   // A, clamped
    return ldwf(k < K ? k : K - 1);
  };
  auto ldx1 = [&](int k, int col, int cen, int nb) -> float {  // k < Cb
    if (MODE == 3) return Xa[(long)k * lda + nb] - Xa[(long)k * lda + cen];
    if (MODE == 4) { int c = k & 127, tau = k >> 7;
                     int t = col + tau - 2;
                     t = t < 0 ? 0 : (t >= ncols ? ncols - 1 : t);
                     return Xa[(long)c * lda + t]; }
    return Xa[(long)k * lda + col];
  };
  auto ldx2 = [&](int k, int col, int colb, int cen) -> float { // Cb <= k < K
    if (MODE == 1) return Xb[(long)(k - Ca) * ldb + col];
    if (MODE == 2) return Xb[(long)(k - Ca) * ldb + colb];
    if (MODE == 3) return Xa[(long)(k - Ca) * lda + cen];
    return 0.f;
  };
  auto ldxg = [&](int k, int col, int colb, int cen, int nb) -> float {
    float r;
    if (MODE == 0 || MODE == 4) {
      r = ldx1(k < K ? k : K - 1, col, cen, nb);
    } else {
      int  k1 = k < Cb ? k : Cb - 1;
      float v1 = ldx1(k1, col, cen, nb);
      int  k2 = k < Cb ? Cb : (k < K ? k : K - 1);
      float v2 = ldx2(k2, col, colb, cen);
      r = (k < Cb) ? v1 : v2;
    }
    return (k < K) ? r : 0.f;
  };

  v8f acc0 = {};
  v8f acc1 = {};
  auto mm = [&](v2f A, v2f B0, v2f B1) {
    acc0 = __builtin_amdgcn_wmma_f32_16x16x4_f32(false, A, false, B0, (short)0, acc0, false, false);
    acc1 = __builtin_amdgcn_wmma_f32_16x16x4_f32(false, A, false, B1, (short)0, acc1, false, false);
  };

  int k0 = 0;
  // region 1: pure first-source loads, no guards at all
  for (; k0 + 4 <= Cb; k0 += 4) {
    int ka = k0 + kh, kb = ka + 1;
    v2f A  = { ldwf(ka), ldwf(kb) };
    v2f B0 = { ldx1(ka, col0, cen0, nb0), ldx1(kb, col0, cen0, nb0) };
    v2f B1 = { ldx1(ka, col1, cen1, nb1), ldx1(kb, col1, cen1, nb1) };
    mm(A, B0, B1);
  }
  if (MODE != 0 && MODE != 4) {
    // straddle iteration across the concat boundary (branchless selects)
    if (k0 < Cb) {
      int ka = k0 + kh, kb = ka + 1;
      v2f A  = { ldw(ka), ldw(kb) };
      v2f B0 = { ldxg(ka, col0, cb0, cen0, nb0), ldxg(kb, col0, cb0, cen0, nb0) };
      v2f B1 = { ldxg(ka, col1, cb1, cen1, nb1), ldxg(kb, col1, cb1, cen1, nb1) };
      mm(A, B0, B1);
      k0 += 4;
    }
    // region 2: pure second-source loads, no guards
    for (; k0 + 4 <= K; k0 += 4) {
      int ka = k0 + kh, kb = ka + 1;
      v2f A  = { ldwf(ka), ldwf(kb) };
      v2f B0 = { ldx2(ka, col0, cb0, cen0), ldx2(kb, col0, cb0, cen0) };
      v2f B1 = { ldx2(ka, col1, cb1, cen1), ldx2(kb, col1, cb1, cen1) };
      mm(A, B0, B1);
    }
  }
  // tail: zero-filled branchless iteration
  if (k0 < K) {
    int ka = k0 + kh, kb = ka + 1;
    v2f A  = { ldw(ka), ldw(kb) };
    v2f B0 = { ldxg(ka, col0, cb0, cen0, nb0), ldxg(kb, col0, cb0, cen0, nb0) };
    v2f B1 = { ldxg(ka, col1, cb1, cen1, nb1), ldxg(kb, col1, cb1, cen1, nb1) };
    mm(A, B0, B1);
  }

  const int rbase = m0 + ((lane >> 4) << 3);
  #pragma unroll
  for (int r = 0; r < 8; ++r) {
    int rr = rbase + r;
    if (rr < O) {
      Y[(long)rr * ncols + col0] = acc0[r];
      Y[(long)rr * ncols + col1] = acc1[r];
    }
  }
}

// ---------------------------------------------------------------------------
// BatchNorm (training stats over B x ncols per channel): one block / channel.
// Emits fused per-channel scale/shift so the apply pass is one FMA + gelu.
// ---------------------------------------------------------------------------
__global__ __launch_bounds__(256)
void bn_stats_k(const float* __restrict__ Y, long sB, int ncols, int B,
                const float* __restrict__ g, const float* __restrict__ be,
                float* __restrict__ scale, float* __restrict__ shift)
{
  const int c = blockIdx.x;
  float s = 0.f, sq = 0.f;
  for (int bb = 0; bb < B; ++bb) {
    const float* p = Y + (long)bb * sB + (long)c * ncols;
    for (int t = threadIdx.x; t < ncols; t += 256) {
      float v = p[t];
      s += v; sq += v * v;
    }
  }
  __shared__ float sh0[256], sh1[256];
  sh0[threadIdx.x] = s; sh1[threadIdx.x] = sq;
  __syncthreads();
  for (int st = 128; st > 0; st >>= 1) {
    if ((int)threadIdx.x < st) {
      sh0[threadIdx.x] += sh0[threadIdx.x + st];
      sh1[threadIdx.x] += sh1[threadIdx.x + st];
    }
    __syncthreads();
  }
  if (threadIdx.x == 0) {
    float n   = (float)((long)B * ncols);
    float m   = sh0[0] / n;
    float var = sh1[0] / n - m * m;
    var = var > 0.f ? var : 0.f;
    float sc = g[c] * rsqrtf(var + 1e-5f);
    scale[c] = sc;
    shift[c] = be[c] - m * sc;
  }
}

// grid = (chunks, C, B); no divisions in the element loop
__global__ __launch_bounds__(256)
void bn_act_k(const float* __restrict__ X, float* __restrict__ Yo,
              const float* __restrict__ scale, const float* __restrict__ shift,
              int ncols)
{
  const int c = blockIdx.y;
  const long base = ((long)blockIdx.z * gridDim.y + c) * ncols;
  const float sc = scale[c], sh = shift[c];
  for (int i = blockIdx.x * 256 + threadIdx.x; i < ncols; i += gridDim.x * 256) {
    float x = X[base + i] * sc + sh;
    float u = 0.7978845608028654f * (x + 0.044715f * x * x * x);  // tanh-gelu
    Yo[base + i] = 0.5f * x * (1.f + tanhf(u));
  }
}

// out[i] = max_j X[i*k + j]   (max over neighbors / stroke points)
__global__ __launch_bounds__(256)
void maxlast_k(const float* __restrict__ X, float* __restrict__ Yo, int k, long total)
{
  long i = (long)blockIdx.x * blockDim.x + threadIdx.x;
  if (i >= total) return;
  const float* p = X + i * k;
  float m = p[0];
  for (int j = 1; j < k; ++j) m = fmaxf(m, p[j]);
  Yo[i] = m;
}

// ---------------------------------------------------------------------------
// kNN over columns of X [C, N]: one block per (b,n). Compile-time k keeps the
// per-thread sorted top-k lists in registers; merge uses lexicographic
// (dist, idx) tie-break (matches stability of top_k(-d)).
// ---------------------------------------------------------------------------
template<int KK>
__global__ __launch_bounds__(256)
void knn_k(const float* __restrict__ X, long sX, int C, int N,
           int* __restrict__ out)
{
  const int n = blockIdx.x;
  const int b = blockIdx.y;
  X += (long)b * sX;
  __shared__ float xc[384];
  __shared__ float cd[256 * KK];
  __shared__ int   ci[256 * KK];
  __shared__ int   cur[256];
  for (int c = threadIdx.x; c < C; c += 256) xc[c] = X[(long)c * N + n];
  __syncthreads();

  float bd[KK]; int bi[KK];
  #pragma unroll
  for (int j = 0; j < KK; ++j) { bd[j] = 3.4e38f; bi[j] = 0x7fffffff; }
  for (int m = threadIdx.x; m < N; m += 256) {
    float d = 0.f;
    for (int c = 0; c < C; ++c) { float t = X[(long)c * N + m] - xc[c]; d += t * t; }
    if (d < bd[KK - 1] || (d == bd[KK - 1] && m < bi[KK - 1])) {
      int j = KK - 1;
      while (j > 0 && (d < bd[j - 1] || (d == bd[j - 1] && m < bi[j - 1]))) {
        bd[j] = bd[j - 1]; bi[j] = bi[j - 1]; --j;
      }
      bd[j] = d; bi[j] = m;
    }
  }
  #pragma unroll
  for (int j = 0; j < KK; ++j) {
    cd[threadIdx.x * KK + j] = bd[j];
    ci[threadIdx.x * KK + j] = bi[j];
  }
  cur[threadIdx.x] = 0;
  __syncthreads();
  if (threadIdx.x == 0) {
    for (int r = 0; r < KK; ++r) {
      float best = 3.5e38f; int besti = 0x7fffffff; int bt = 0;
      for (int t = 0; t < 256; ++t) {
        if (cur[t] < KK) {
          float d = cd[t * KK + cur[t]]; int ii = ci[t * KK + cur[t]];
          if (d < best || (d == best && ii < besti)) { best = d; besti = ii; bt = t; }
        }
      }
      cur[bt]++;
      out[((long)b * N + n) * KK + r] = besti;
    }
  }
}

// ---------------------------------------------------------------------------
// Host-side orchestration
// ---------------------------------------------------------------------------
template<int MODE>
static inline void gemm_launch(hipStream_t st, const float* W,
                               const float* Xa, long sXa, int lda,
                               const float* Xb, long sXb, int ldb,
                               const int* nbr, long sNbr,
                               float* Y, int O, int K, int Ca, int ncols,
                               int knn, int inner)
{
  dim3 blk(32, 4, 1);
  dim3 grd((ncols + 127) / 128, (O + 15) / 16, 8);
  gemm_wmma_k<MODE><<<grd, blk, 0, st>>>(W, Xa, sXa, lda, Xb, sXb, ldb, nbr, sNbr,
                                         Y, (long)O * ncols, O, K, Ca, ncols, knn, inner);
}

static inline void bn_launch(hipStream_t st, const float* X, float* Yo,
                             const float* g, const float* be,
                             int C, int ncols, float* scale, float* shift)
{
  bn_stats_k<<<C, 256, 0, st>>>(X, (long)C * ncols, ncols, 8, g, be, scale, shift);
  int nx = (ncols + 255) / 256; if (nx > 80) nx = 80;
  bn_act_k<<<dim3(nx, C, 8), 256, 0, st>>>(X, Yo, scale, shift, ncols);
}

static inline void max_launch(hipStream_t st, const float* X, float* Yo, int k, long total)
{
  maxlast_k<<<(int)((total + 255) / 256), 256, 0, st>>>(X, Yo, k, total);
}

template<int KK>
static inline void knn_launch(hipStream_t st, const float* X, int C, int N, int* out)
{
  knn_k<KK><<<dim3(N, 8), 256, 0, st>>>(X, (long)C * N, C, N, out);
}

extern "C" void kernel_launch(void* const* d_in, const int* in_sizes, int n_in,
                              void* d_out, int out_size, void* d_ws, size_t ws_size,
                              hipStream_t stream)
{
  (void)in_sizes; (void)n_in; (void)out_size; (void)ws_size;
  auto in = [&](int i) { return (const float*)d_in[i]; };

  const float* sparse = in(0);                 // [8,256,32]
  const float* dense  = in(1);                 // [8,128,32,64] == [8,128,2048]

  // ---- workspace carve-out (floats) ----
  float* ws = (float*)d_ws;
  size_t off = 0;
  auto alloc = [&](size_t n) { float* p = ws + off; off += n; return p; };
  float* big1  = alloc(49807360ul);            // up to [8,304,20480]
  float* big2  = alloc(41943040ul);            // up to [8,256,20480]
  float* buf_t = alloc(8ul * 128 * 2048);      // temporal conv out
  float* sp_dn = alloc(8ul * 128 * 32);
  float* us0   = alloc(8ul * 256 * 32);        // union_sparse after enc
  float* ud0   = alloc(8ul * 128 * 2048);      // union_dense after enc
  float* x1s   = alloc(8ul * 362 * 32);
  float* x2s   = alloc(8ul * 512 * 32);
  float* x1d   = alloc(8ul * 181 * 2048);
  float* x2d   = alloc(8ul * 256 * 2048);
  float* ta    = alloc(8ul * 668 * 64);        // sparse-branch temporaries
  float* tb    = alloc(8ul * 512 * 64);
  float* scale = alloc(1024);
  float* shift = alloc(1024);
  int*   idxS  = (int*)alloc(8ul * 32 * 2);
  int*   idxD  = (int*)alloc(8ul * 2048 * 10);

  float* outS = (float*)d_out;                 // [8,512,32]
  float* outD = outS + 8 * 512 * 32;           // [8,256,2048]

  // ===== DenseToSparse: temporal conv(k=5, edge) -> BN/gelu -> max over pts
  gemm_launch<4>(stream, in(2), dense, 128L * 2048, 2048, nullptr, 0, 0, nullptr, 0,
                 buf_t, 128, 640, 128, 2048, 0, 0);
  bn_launch(stream, buf_t, buf_t, in(4), in(5), 128, 2048, scale, shift);
  max_launch(stream, buf_t, sp_dn, 64, 8L * 128 * 32);
  // d2s_enc: concat(sparse 256, sp_from_dn 128) -> 256
  gemm_launch<1>(stream, in(6), sparse, 256L * 32, 32, sp_dn, 128L * 32, 32, nullptr, 0,
                 us0, 256, 384, 256, 32, 0, 0);
  bn_launch(stream, us0, us0, in(8), in(9), 256, 32, scale, shift);

  // ===== SparseToDense: concat(dense 128, broadcast(sparse) 256) -> 128
  gemm_launch<2>(stream, in(10), dense, 128L * 2048, 2048, sparse, 256L * 32, 32, nullptr, 0,
                 ud0, 128, 384, 128, 2048, 0, 64);
  bn_launch(stream, ud0, ud0, in(12), in(13), 128, 2048, scale, shift);

  // ===== SparseUpdate (DGCNN, N=32, k=2) =====
  knn_launch<2>(stream, us0, 256, 32, idxS);
  gemm_launch<3>(stream, in(14), us0, 256L * 32, 32, nullptr, 0, 0, idxS, 64,
                 ta, 430, 512, 256, 64, 2, 0);
  bn_launch(stream, ta, ta, in(16), in(17), 430, 64, scale, shift);
  gemm_launch<0>(stream, in(18), ta, 430L * 64, 64, nullptr, 0, 0, nullptr, 0,
                 tb, 362, 430, 0, 64, 0, 0);
  bn_launch(stream, tb, tb, in(20), in(21), 362, 64, scale, shift);
  max_launch(stream, tb, x1s, 2, 8L * 362 * 32);

  knn_launch<2>(stream, x1s, 362, 32, idxS);
  gemm_launch<3>(stream, in(22), x1s, 362L * 32, 32, nullptr, 0, 0, idxS, 64,
                 ta, 608, 724, 362, 64, 2, 0);
  bn_launch(stream, ta, ta, in(24), in(25), 608, 64, scale, shift);
  gemm_launch<0>(stream, in(26), ta, 608L * 64, 64, nullptr, 0, 0, nullptr, 0,
                 tb, 512, 608, 0, 64, 0, 0);
  bn_launch(stream, tb, tb, in(28), in(29), 512, 64, scale, shift);
  max_launch(stream, tb, x2s, 2, 8L * 512 * 32);

  gemm_launch<1>(stream, in(30), x1s, 362L * 32, 32, x2s, 512L * 32, 32, nullptr, 0,
                 ta, 668, 874, 362, 32, 0, 0);
  bn_launch(stream, ta, ta, in(32), in(33), 668, 32, scale, shift);
  gemm_launch<0>(stream, in(34), ta, 668L * 32, 32, nullptr, 0, 0, nullptr, 0,
                 tb, 512, 668, 0, 32, 0, 0);
  bn_launch(stream, tb, outS, in(36), in(37), 512, 32, scale, shift);

  // ===== DenseUpdate (DGCNN, N=2048, k=10) =====
  knn_launch<10>(stream, ud0, 128, 2048, idxD);
  gemm_launch<3>(stream, in(38), ud0, 128L * 2048, 2048, nullptr, 0, 0, idxD, 20480,
                 big1, 215, 256, 128, 20480, 10, 0);
  bn_launch(stream, big1, big1, in(40), in(41), 215, 20480, scale, shift);
  gemm_launch<0>(stream, in(42), big1, 215L * 20480, 20480, nullptr, 0, 0, nullptr, 0,
                 big2, 181, 215, 0, 20480, 0, 0);
  bn_launch(stream, big2, big2, in(44), in(45), 181, 20480, scale, shift);
  max_launch(stream, big2, x1d, 10, 8L * 181 * 2048);

  knn_launch<10>(stream, x1d, 181, 2048, idxD);
  gemm_launch<3>(stream, in(46), x1d, 181L * 2048, 2048, nullptr, 0, 0, idxD, 20480,
                 big1, 304, 362, 181, 20480, 10, 0);
  bn_launch(stream, big1, big1, in(48), in(49), 304, 20480, scale, shift);
  gemm_launch<0>(stream, in(50), big1, 304L * 20480, 20480, nullptr, 0, 0, nullptr, 0,
                 big2, 256, 304, 0, 20480, 0, 0);
  bn_launch(stream, big2, big2, in(52), in(53), 256, 20480, scale, shift);
  max_launch(stream, big2, x2d, 10, 8L * 256 * 2048);

  gemm_launch<1>(stream, in(54), x1d, 181L * 2048, 2048, x2d, 256L * 2048, 2048, nullptr, 0,
                 big1, 334, 437, 181, 2048, 0, 0);
  bn_launch(stream, big1, big1, in(56), in(57), 334, 2048, scale, shift);
  gemm_launch<0>(stream, in(58), big1, 334L * 2048, 2048, nullptr, 0, 0, nullptr, 0,
                 big2, 256, 334, 0, 2048, 0, 0);
  bn_launch(stream, big2, outD, in(60), in(61), 256, 2048, scale, shift);
}